// DiscriminativeLoss_54125177864479
// MI455X (gfx1250) — compile-verified
//
#include <hip/hip_runtime.h>
#include <hip/hip_bf16.h>

// Discriminative loss for MI455X (gfx1250).
//
// Roofline: 80 MB of inputs, two passes (means needed before pull pass), all
// L2-resident (192 MB L2).  HBM floor ~3.5us @ 23.3 TB/s => bandwidth-bound.
// f32 end-to-end.  Streaming uses global_load_b128 (float4: 4 pixels/thread)
// per channel plane for minimal VMEM issue at full coalescing.  Reductions:
// per-wave LDS bins (ds_add_f32) -> fixed-order cross-wave -> one
// global_atomic_add_f32 per bin per block.  Push term = per-batch 16x8 mean
// Gram via V_WMMA_F32_16X16X4_F32 (segments 1..16 = the 16 WMMA rows).

#define NB    8
#define CCH   8
#define NPIX  (512 * 512)
#define KSEG  16                   // foreground segments 1..16 -> rows 0..15
#define NCOL  9                    // 8 channel sums + count
#define BPB   64                   // blocks per batch (512 blocks total)
#define CHUNK (NPIX / BPB)         // 4096 pixels per block
#define VPT   4                    // pixels per thread per iteration (float4)
#define ITERS (CHUNK / (256 * VPT))  // 4

typedef __attribute__((ext_vector_type(2))) float v2f;
typedef __attribute__((ext_vector_type(8))) float v8f;

__device__ __forceinline__ float f4get(const float4& v, int j) {
    return j == 0 ? v.x : (j == 1 ? v.y : (j == 2 ? v.z : v.w));
}
__device__ __forceinline__ int i4get(const int4& v, int j) {
    return j == 0 ? v.x : (j == 1 ? v.y : (j == 2 ? v.z : v.w));
}

// ---------------------------------------------------------------- zero ws ---
__global__ void zero_ws(float* __restrict__ ws, int n) {
    int i = blockIdx.x * blockDim.x + threadIdx.x;
    if (i < n) ws[i] = 0.0f;
}

// ------------------------------------------------- pass 1: segment sums -----
// bins[b][s0][c] : c=0..7 channel sums, c=8 pixel count (exact in f32: <2^24).
__global__ __launch_bounds__(256) void pass1(const float* __restrict__ emb,
                                             const int*   __restrict__ lab,
                                             const int*   __restrict__ msk,
                                             float*       __restrict__ bins) {
    __shared__ float wb[8][KSEG][NCOL];       // per-wave bins (8 wave32s)
    const int t    = threadIdx.x;
    const int wave = t >> 5;

    for (int i = t; i < 8 * KSEG * NCOL; i += 256) ((float*)wb)[i] = 0.0f;
    __syncthreads();

    const int blk   = blockIdx.x;
    const int b     = blk / BPB;
    const int chunk = blk % BPB;
    const int qbase = chunk * (CHUNK / VPT);   // quad (float4) index base

    const float* eb = emb + (size_t)b * CCH * NPIX;
    const int4*  lb = (const int4*)(lab + (size_t)b * NPIX);
    const int4*  mb = (const int4*)(msk + (size_t)b * NPIX);

    for (int i = 0; i < ITERS; ++i) {
        const int q = qbase + i * 256 + t;     // quad index; pixels 4q..4q+3
        // gfx1250 global_prefetch_b8: next iteration's channel-0 line
        __builtin_prefetch((const float4*)eb + q + 256, 0, 3);

        const int4 l4 = lb[q];
        const int4 m4 = mb[q];
        int s[VPT];
#pragma unroll
        for (int j = 0; j < VPT; ++j) s[j] = i4get(l4, j) * i4get(m4, j);

        if ((s[0] | s[1] | s[2] | s[3]) > 0) { // any foreground in this quad
            float4 e4[CCH];
#pragma unroll
            for (int c = 0; c < CCH; ++c)
                e4[c] = ((const float4*)(eb + (size_t)c * NPIX))[q];   // b128
#pragma unroll
            for (int j = 0; j < VPT; ++j) {
                if (s[j] > 0) {
                    const int s0 = s[j] - 1;
#pragma unroll
                    for (int c = 0; c < CCH; ++c)
                        atomicAdd(&wb[wave][s0][c], f4get(e4[c], j)); // ds_add_f32
                    atomicAdd(&wb[wave][s0][8], 1.0f);
                }
            }
        }
    }
    __syncthreads();

    // fixed-order cross-wave reduce, then one global atomic per bin
    if (t < KSEG * NCOL) {
        const int s0 = t / NCOL, c = t % NCOL;
        float acc = 0.0f;
#pragma unroll
        for (int w = 0; w < 8; ++w) acc += wb[w][s0][c];
        atomicAdd(&bins[(b * KSEG + s0) * NCOL + c], acc); // global_atomic_add_f32
    }
}

// -------------------------------------------- pass 2: pull penalties --------
__global__ __launch_bounds__(256) void pass2(const float* __restrict__ emb,
                                             const int*   __restrict__ lab,
                                             const int*   __restrict__ msk,
                                             const float* __restrict__ bins,
                                             float*       __restrict__ penS) {
    __shared__ float mean[KSEG][CCH];
    __shared__ float pw[8][KSEG];
    const int t    = threadIdx.x;
    const int wave = t >> 5;

    const int blk   = blockIdx.x;
    const int b     = blk / BPB;
    const int chunk = blk % BPB;

    for (int i = t; i < KSEG * CCH; i += 256) {
        const int s0 = i >> 3, c = i & 7;
        const float cnt = bins[(b * KSEG + s0) * NCOL + 8];
        mean[s0][c] = bins[(b * KSEG + s0) * NCOL + c] / fmaxf(cnt, 1.0f);
    }
    for (int i = t; i < 8 * KSEG; i += 256) ((float*)pw)[i] = 0.0f;
    __syncthreads();

    const int qbase = chunk * (CHUNK / VPT);
    const float* eb = emb + (size_t)b * CCH * NPIX;
    const int4*  lb = (const int4*)(lab + (size_t)b * NPIX);
    const int4*  mb = (const int4*)(msk + (size_t)b * NPIX);

    for (int i = 0; i < ITERS; ++i) {
        const int q = qbase + i * 256 + t;
        const int4 l4 = lb[q];
        const int4 m4 = mb[q];
        int s[VPT];
#pragma unroll
        for (int j = 0; j < VPT; ++j) s[j] = i4get(l4, j) * i4get(m4, j);

        if ((s[0] | s[1] | s[2] | s[3]) > 0) {
            float4 e4[CCH];
#pragma unroll
            for (int c = 0; c < CCH; ++c)
                e4[c] = ((const float4*)(eb + (size_t)c * NPIX))[q];
#pragma unroll
            for (int j = 0; j < VPT; ++j) {
                if (s[j] > 0) {
                    const int s0 = s[j] - 1;
                    float d2 = 0.0f;
#pragma unroll
                    for (int c = 0; c < CCH; ++c) {
                        const float diff = f4get(e4[c], j) - mean[s0][c];
                        d2 = fmaf(diff, diff, d2);
                    }
                    const float d = sqrtf(fmaxf(d2, 1e-12f));
                    const float h = fmaxf(d - 0.5f, 0.0f);   // DELTA_V = 0.5
                    atomicAdd(&pw[wave][s0], h * h);
                }
            }
        }
    }
    __syncthreads();

    if (t < KSEG) {
        float acc = 0.0f;
#pragma unroll
        for (int w = 0; w < 8; ++w) acc += pw[w][t];
        atomicAdd(&penS[b * KSEG + t], acc);
    }
}

// ---------------------------- finalize: pull average + WMMA Gram push -------
__global__ __launch_bounds__(256) void finalize(const float* __restrict__ bins,
                                                const float* __restrict__ penS,
                                                float*       __restrict__ out) {
    __shared__ float mu[NB][KSEG][CCH];
    __shared__ float nrm[NB][KSEG];
    __shared__ float prs[NB][KSEG];      // present flag as float
    __shared__ float pullb[NB], pushb[NB], Kb[NB];
    __shared__ float hacc[NB][32];

    const int t    = threadIdx.x;
    const int wave = t >> 5;             // wave == batch
    const int lane = t & 31;

    // --- means, norms, counts ---
    for (int i = t; i < NB * KSEG; i += 256) {
        const int b = i >> 4, s0 = i & 15;
        const float cnt = bins[(b * KSEG + s0) * NCOL + 8];
        prs[b][s0] = (cnt > 0.0f) ? 1.0f : 0.0f;
        float n2 = 0.0f;
#pragma unroll
        for (int c = 0; c < CCH; ++c) {
            const float m = bins[(b * KSEG + s0) * NCOL + c] / fmaxf(cnt, 1.0f);
            mu[b][s0][c] = m;
            n2 = fmaf(m, m, n2);
        }
        nrm[b][s0] = n2;
    }
    __syncthreads();

    // --- pull term per batch ---
    if (t < NB) {
        const int b = t;
        float K = 0.0f, psum = 0.0f;
        for (int s0 = 0; s0 < KSEG; ++s0) {
            const float cnt = bins[(b * KSEG + s0) * NCOL + 8];
            if (cnt > 0.0f) {
                K += 1.0f;
                psum += penS[b * KSEG + s0] / cnt;  // cnt>=1 => max(cnt,1)==cnt
            }
        }
        Kb[b] = K;
        pullb[b] = psum / fmaxf(K, 1.0f);
    }
    __syncthreads();

    // --- push term: G = MU * MU^T via V_WMMA_F32_16X16X4_F32 (2 K-chunks) ---
    // f32 A layout (16x4): lane = h*16 + r, r = row M; VGPR0 -> K = 2h,
    // VGPR1 -> K = 2h+1.  Symmetric Gram => B operand registers == A operand.
    const int b = wave;
    const int r = lane & 15, h = lane >> 4;
    v2f a0, a1;
    a0.x = mu[b][r][2 * h + 0];
    a0.y = mu[b][r][2 * h + 1];
    a1.x = mu[b][r][4 + 2 * h + 0];
    a1.y = mu[b][r][4 + 2 * h + 1];

    v8f g = {};
    g = __builtin_amdgcn_wmma_f32_16x16x4_f32(false, a0, false, a0,
                                              (short)0, g, false, false);
    g = __builtin_amdgcn_wmma_f32_16x16x4_f32(false, a1, false, a1,
                                              (short)0, g, false, false);

    // C/D layout: VGPR j -> M = j + 8h, N = lane%16.  Branch-free hinge.
    float hs = 0.0f;
#pragma unroll
    for (int j = 0; j < 8; ++j) {
        const int M = j + 8 * h;
        const int N = r;
        const float pm = (M < N) ? prs[b][M] * prs[b][N] : 0.0f;
        const float d2 = nrm[b][M] + nrm[b][N] - 2.0f * g[j];
        const float d  = sqrtf(fmaxf(d2, 1e-12f));
        const float x  = fmaxf(3.0f - d, 0.0f);     // 2*DELTA_D = 3.0
        hs = fmaf(pm, x * x, hs);
    }
    hacc[wave][lane] = hs;
    __syncthreads();

    if (t < NB) {
        const int bb = t;
        float s = 0.0f;
        for (int l = 0; l < 32; ++l) s += hacc[bb][l];
        const float K  = Kb[bb];
        const float np = K * (K - 1.0f) * 0.5f;     // #pairs i<j both present
        pushb[bb] = s / fmaxf(np, 1.0f);
    }
    __syncthreads();

    if (t == 0) {
        float nv = 0.0f, lp = 0.0f, lpu = 0.0f;
        for (int bb = 0; bb < NB; ++bb) {
            const float v = (Kb[bb] > 0.0f) ? 1.0f : 0.0f;
            nv += v;
            lp += pullb[bb] * v;
            lpu += pushb[bb] * v;
        }
        nv = fmaxf(nv, 1.0f);
        out[0] = lp / nv;    // loss_pull
        out[1] = lpu / nv;   // loss_push
    }
}

// ---------------------------------------------------------------------------
extern "C" void kernel_launch(void* const* d_in, const int* in_sizes, int n_in,
                              void* d_out, int out_size, void* d_ws, size_t ws_size,
                              hipStream_t stream) {
    const float* emb = (const float*)d_in[0];   // (8,8,512,512) f32
    const int*   lab = (const int*)d_in[1];     // (8,512,512) i32
    const int*   msk = (const int*)d_in[2];     // (8,512,512) i32
    float*       out = (float*)d_out;           // 2 floats: (pull, push)

    float* ws   = (float*)d_ws;
    float* bins = ws;                           // NB*KSEG*NCOL = 1152 floats
    float* penS = ws + NB * KSEG * NCOL;        // NB*KSEG     =  128 floats
    const int nzero = NB * KSEG * NCOL + NB * KSEG;

    hipLaunchKernelGGL(zero_ws, dim3((nzero + 255) / 256), dim3(256), 0, stream,
                       ws, nzero);
    hipLaunchKernelGGL(pass1, dim3(NB * BPB), dim3(256), 0, stream,
                       emb, lab, msk, bins);
    hipLaunchKernelGGL(pass2, dim3(NB * BPB), dim3(256), 0, stream,
                       emb, lab, msk, bins, penS);
    hipLaunchKernelGGL(finalize, dim3(1), dim3(256), 0, stream,
                       bins, penS, out);
}